// CoherentBiMamba_31628139168143
// MI455X (gfx1250) — compile-verified
//
#include <hip/hip_runtime.h>
#include <hip/hip_bf16.h>

// ---------------------------------------------------------------------------
// CoherentBiMamba on gfx1250 (MI455X).
//  - All dense projections use V_WMMA_F32_16X16X4_F32 (full f32 precision;
//    the workload is latency-bound at ~5 GFLOP total, everything L2-resident,
//    so f32 WMMA is the right precision/path).
//  - GEMM is templated on K so the WMMA K-loop fully unrolls: loads are
//    hoisted/batched by the scheduler, no runtime loop control.
//  - Out-of-range columns are handled by index clamping (garbage flows only
//    into never-stored output columns) -> zero EXEC masking in the hot loop.
//  - Selective scan parallelized as a 3-phase chunked scan (64 chunks x 64
//    steps) using prod_t exp(dt_t*A) == exp(A * sum_t dt_t).
//  - Backward direction handled purely by index reversal.
// ---------------------------------------------------------------------------

#define BSZ     2
#define LSEQ    4096
#define DMODEL  64
#define DINNER  128
#define DSTATE  16
#define DTRANK  4
#define NLAYERS 4
#define NCLS    4
#define ROWS    (BSZ * LSEQ)      // 8192
#define XDLD    48                // padded leading dim for x_dbl (36 -> 48)
#define NCHUNK  64
#define CLEN    64                // NCHUNK * CLEN == LSEQ

typedef __attribute__((ext_vector_type(2))) float v2f;
typedef __attribute__((ext_vector_type(8))) float v8f;

__device__ __forceinline__ float softplus_f(float x) {
    return (x > 20.f) ? x : log1pf(__expf(x));
}
__device__ __forceinline__ float silu_f(float x) {
    return x / (1.f + __expf(-x));
}

// ---------------------------------------------------------------------------
// Wave-tile f32 WMMA GEMM:  C[M,N](ldc) (+)= A[M,K](lda) * W[N,K]^T (+bias)(+act)
// One 16x16 output tile per wave; fully unrolled chain of
// V_WMMA_F32_16X16X4_F32 (K/4 instructions).
// Fragment layouts per CDNA5 ISA 7.12.2:
//   A (16x4):  lanes 0-15 -> M=lane, v0=K0 v1=K1 ; lanes 16-31 -> v0=K2 v1=K3
//   B (4x16):  rows striped across lanes within a VGPR (mirror of C/D)
//   C/D:       vgpr r: lanes 0-15 -> M=r, lanes 16-31 -> M=r+8, N=lane&15
// ---------------------------------------------------------------------------
template <int K, int ACT, int ACCUM>
__global__ __launch_bounds__(128) void gemm_wmma_f32(
    const float* __restrict__ A, int lda,
    const float* __restrict__ W,
    const float* __restrict__ bias,
    float* __restrict__ C, int ldc,
    int M, int N)
{
    const int lane = threadIdx.x & 31;
    const int wv   = threadIdx.x >> 5;
    const int tilesN = (N + 15) >> 4;
    const int tilesM = M >> 4;
    const int tile = blockIdx.x * 4 + wv;
    if (tile >= tilesM * tilesN) return;     // wave-uniform; EXEC all-1 inside

    const int tm = tile / tilesN;
    const int tn = tile - tm * tilesN;
    const int m0 = tm << 4, n0 = tn << 4;
    const int l15 = lane & 15;
    const int hi  = lane >> 4;               // 0: K0/K1 half, 1: K2/K3 half
    const int rowA = m0 + l15;
    const int col  = n0 + l15;
    const bool colok = (col < N);
    // Clamp instead of predicate: garbage columns are never stored.
    const int colc = colok ? col : (N - 1);

    const float* __restrict__ Arow = A + rowA * lda + hi * 2;
    const float* __restrict__ Wrow = W + colc * K + hi * 2;

    v8f c;
#pragma unroll
    for (int r = 0; r < 8; ++r) c[r] = 0.f;
    if (ACCUM) {
#pragma unroll
        for (int r = 0; r < 8; ++r)
            c[r] = C[(m0 + r + hi * 8) * ldc + colc];
    }

    // Preload fragments (8-byte aligned for all K/lda used here), then the
    // fully unrolled WMMA chain.
    v2f afrag[K / 4], bfrag[K / 4];
#pragma unroll
    for (int kk = 0; kk < K / 4; ++kk) {
        afrag[kk] = *(const v2f*)(Arow + kk * 4);
        bfrag[kk] = *(const v2f*)(Wrow + kk * 4);
    }
#pragma unroll
    for (int kk = 0; kk < K / 4; ++kk) {
        c = __builtin_amdgcn_wmma_f32_16x16x4_f32(
                /*neg_a=*/false, afrag[kk], /*neg_b=*/false, bfrag[kk],
                /*c_mod=*/(short)0, c, /*reuse_a=*/false, /*reuse_b=*/false);
    }

    if (colok) {
        const float bv = bias ? bias[col] : 0.f;
#pragma unroll
        for (int r = 0; r < 8; ++r) {
            int row = m0 + r + hi * 8;
            float v = c[r] + bv;
            if (ACT == 1) v = softplus_f(v);
            C[row * ldc + col] = v;
        }
    }
}

// ---------------------------------------------------------------------------
// h[b,t,dm] = in_w[dm,0]*x[b,0,t] + in_w[dm,1]*x[b,1,t] + in_b[dm]
// ---------------------------------------------------------------------------
__global__ __launch_bounds__(256) void embed_kernel(
    const float* __restrict__ x, const float* __restrict__ inw,
    const float* __restrict__ inb, float* __restrict__ h)
{
    int idx = blockIdx.x * blockDim.x + threadIdx.x;
    if (idx >= ROWS * DMODEL) return;
    int dm = idx & 63;
    int rest = idx >> 6;
    int t = rest & (LSEQ - 1);
    int b = rest >> 12;
    h[(b * LSEQ + t) * DMODEL + dm] =
        inw[dm * 2 + 0] * x[(b * 2 + 0) * LSEQ + t] +
        inw[dm * 2 + 1] * x[(b * 2 + 1) * LSEQ + t] + inb[dm];
}

// ---------------------------------------------------------------------------
// Depthwise causal conv (K=4) + SiLU.  u input lives in xz[:, 0:128].
// rev=1 processes the time-reversed sequence without materializing a flip.
// ---------------------------------------------------------------------------
__global__ __launch_bounds__(256) void conv_silu_kernel(
    const float* __restrict__ xz, const float* __restrict__ cw,
    const float* __restrict__ cb, float* __restrict__ u, int rev)
{
    int idx = blockIdx.x * blockDim.x + threadIdx.x;
    if (idx >= ROWS * DINNER) return;
    int d = idx & 127;
    int rest = idx >> 7;
    int p = rest & (LSEQ - 1);
    int b = rest >> 12;
    float acc = cb[d];
#pragma unroll
    for (int k = 0; k < 4; ++k) {
        int off = 3 - k;
        int q = rev ? (p + off) : (p - off);
        if (q >= 0 && q < LSEQ)
            acc += cw[d * 4 + k] * xz[(b * LSEQ + q) * 256 + d];
    }
    u[(b * LSEQ + p) * DINNER + d] = silu_f(acc);
}

// ---------------------------------------------------------------------------
// Scan phase 1: per-chunk local scan from h=0. Stores chunk-final state Hend
// and chunk dt-sum (prod exp(dt*A) == exp(A * sum dt)).
// Thread = (b, d, chunk); d in low bits for coalesced dt/u loads.
// ---------------------------------------------------------------------------
__global__ __launch_bounds__(128) void scan_phase1(
    const float* __restrict__ dt, const float* __restrict__ u,
    const float* __restrict__ xdbl, const float* __restrict__ A_log,
    float* __restrict__ Hend, float* __restrict__ dtsum, int rev)
{
    int tid = blockIdx.x * blockDim.x + threadIdx.x;
    if (tid >= BSZ * DINNER * NCHUNK) return;
    int d = tid & 127;
    int bc = tid >> 7;
    int b = bc & 1;
    int c = bc >> 1;

    float Adn[DSTATE];
#pragma unroll
    for (int n = 0; n < DSTATE; ++n) Adn[n] = -__expf(A_log[d * DSTATE + n]);

    float h[DSTATE];
#pragma unroll
    for (int n = 0; n < DSTATE; ++n) h[n] = 0.f;
    float dts = 0.f;

    const int t0 = c * CLEN;
    for (int i = 0; i < CLEN; ++i) {
        int t = t0 + i;
        int p = rev ? (LSEQ - 1 - t) : t;
        int base = b * LSEQ + p;
        if (i + 1 < CLEN) {
            int pn = rev ? (p - 1) : (p + 1);
            __builtin_prefetch(&dt[(b * LSEQ + pn) * DINNER + d], 0, 0);
        }
        float dtv = dt[base * DINNER + d];
        float uv  = u[base * DINNER + d];
        dts += dtv;
        float dtu = dtv * uv;
        const float* xb = xdbl + base * XDLD;
#pragma unroll
        for (int n = 0; n < DSTATE; ++n)
            h[n] = __expf(dtv * Adn[n]) * h[n] + dtu * xb[DTRANK + n];
    }

    int ho = ((b * DINNER + d) * NCHUNK + c) * DSTATE;
#pragma unroll
    for (int n = 0; n < DSTATE; ++n) Hend[ho + n] = h[n];
    dtsum[(b * DINNER + d) * NCHUNK + c] = dts;
}

// ---------------------------------------------------------------------------
// Scan phase 2: tiny sequential carry across the 64 chunk summaries.
// Thread = (b, d). Hin[c] = carry-in state for chunk c.
// ---------------------------------------------------------------------------
__global__ __launch_bounds__(256) void scan_phase2(
    const float* __restrict__ Hend, const float* __restrict__ dtsum,
    const float* __restrict__ A_log, float* __restrict__ Hin)
{
    int tid = blockIdx.x * blockDim.x + threadIdx.x;
    if (tid >= BSZ * DINNER) return;
    int d = tid & 127;
    int b = tid >> 7;

    float Adn[DSTATE];
#pragma unroll
    for (int n = 0; n < DSTATE; ++n) Adn[n] = -__expf(A_log[d * DSTATE + n]);

    float h[DSTATE];
#pragma unroll
    for (int n = 0; n < DSTATE; ++n) h[n] = 0.f;

    for (int c = 0; c < NCHUNK; ++c) {
        int o = ((b * DINNER + d) * NCHUNK + c) * DSTATE;
        float ds = dtsum[(b * DINNER + d) * NCHUNK + c];
#pragma unroll
        for (int n = 0; n < DSTATE; ++n) {
            Hin[o + n] = h[n];
            h[n] = __expf(Adn[n] * ds) * h[n] + Hend[o + n];
        }
    }
}

// ---------------------------------------------------------------------------
// Scan phase 3: replay each chunk with its carry-in, emit
//   y_out = (sum_n h*C + u*D) * silu(z)
// ---------------------------------------------------------------------------
__global__ __launch_bounds__(128) void scan_phase3(
    const float* __restrict__ dt, const float* __restrict__ u,
    const float* __restrict__ xdbl, const float* __restrict__ xz,
    const float* __restrict__ A_log, const float* __restrict__ Dvec,
    const float* __restrict__ Hin, float* __restrict__ yout, int rev)
{
    int tid = blockIdx.x * blockDim.x + threadIdx.x;
    if (tid >= BSZ * DINNER * NCHUNK) return;
    int d = tid & 127;
    int bc = tid >> 7;
    int b = bc & 1;
    int c = bc >> 1;

    float Adn[DSTATE];
#pragma unroll
    for (int n = 0; n < DSTATE; ++n) Adn[n] = -__expf(A_log[d * DSTATE + n]);

    float h[DSTATE];
    int ho = ((b * DINNER + d) * NCHUNK + c) * DSTATE;
#pragma unroll
    for (int n = 0; n < DSTATE; ++n) h[n] = Hin[ho + n];

    const float Dd = Dvec[d];
    const int t0 = c * CLEN;
    for (int i = 0; i < CLEN; ++i) {
        int t = t0 + i;
        int p = rev ? (LSEQ - 1 - t) : t;
        int base = b * LSEQ + p;
        if (i + 1 < CLEN) {
            int pn = rev ? (p - 1) : (p + 1);
            __builtin_prefetch(&u[(b * LSEQ + pn) * DINNER + d], 0, 0);
        }
        float dtv = dt[base * DINNER + d];
        float uv  = u[base * DINNER + d];
        float dtu = dtv * uv;
        const float* xb = xdbl + base * XDLD;
        float y = 0.f;
#pragma unroll
        for (int n = 0; n < DSTATE; ++n) {
            h[n] = __expf(dtv * Adn[n]) * h[n] + dtu * xb[DTRANK + n];
            y += h[n] * xb[DTRANK + DSTATE + n];
        }
        float z = xz[base * 256 + DINNER + d];
        yout[base * DINNER + d] = (y + uv * Dd) * silu_f(z);
    }
}

// ---------------------------------------------------------------------------
// h = LayerNorm(h + add) * g + b   (one thread per 64-wide row)
// ---------------------------------------------------------------------------
__global__ __launch_bounds__(256) void add_ln_kernel(
    float* __restrict__ h, const float* __restrict__ add,
    const float* __restrict__ g, const float* __restrict__ bb)
{
    int r = blockIdx.x * blockDim.x + threadIdx.x;
    if (r >= ROWS) return;
    float v[DMODEL];
    const float* hr = h + r * DMODEL;
    float mu = 0.f;
#pragma unroll
    for (int i = 0; i < DMODEL; ++i) {
        float a = add ? add[r * DMODEL + i] : 0.f;
        v[i] = hr[i] + a;
        mu += v[i];
    }
    mu *= (1.f / DMODEL);
    float var = 0.f;
#pragma unroll
    for (int i = 0; i < DMODEL; ++i) { float dd = v[i] - mu; var += dd * dd; }
    var *= (1.f / DMODEL);
    float rs = rsqrtf(var + 1e-5f);
    float* ho = h + r * DMODEL;
#pragma unroll
    for (int i = 0; i < DMODEL; ++i)
        ho[i] = (v[i] - mu) * rs * g[i] + bb[i];
}

// ---------------------------------------------------------------------------
// logits (B, NCLS, L):  out[b,c,t] = sum_k h[b,t,k]*head_w[c,k] + head_b[c]
// ---------------------------------------------------------------------------
__global__ __launch_bounds__(256) void head_kernel(
    const float* __restrict__ h, const float* __restrict__ hw,
    const float* __restrict__ hb, float* __restrict__ out)
{
    int idx = blockIdx.x * blockDim.x + threadIdx.x;
    if (idx >= ROWS * NCLS) return;
    int c = idx & 3;
    int rest = idx >> 2;
    int t = rest & (LSEQ - 1);
    int b = rest >> 12;
    float s = hb[c];
#pragma unroll
    for (int k = 0; k < DMODEL; ++k)
        s += h[(b * LSEQ + t) * DMODEL + k] * hw[c * DMODEL + k];
    out[(b * NCLS + c) * LSEQ + t] = s;
}

// ---------------------------------------------------------------------------
// Host-side orchestration.
// Input leaf order assumes the harness flattens setup_inputs() in Python dict
// insertion order, recursing into nested dicts/lists:
//   0:x  1:in_w  2:in_b
//   per layer l (base = 3 + 20*l), per dir (fwd then bwd, +9 each):
//     +0 in_w(256,64) +1 conv_w(128,4) +2 conv_b +3 xproj_w(36,128)
//     +4 dt_w(128,4)  +5 dt_b          +6 A_log(128,16) +7 D +8 out_w(64,128)
//   base+18 ln_g, base+19 ln_b
//   83 norm_g  84 norm_b  85 head_w(4,64)  86 head_b(4)
// ---------------------------------------------------------------------------
extern "C" void kernel_launch(void* const* d_in, const int* in_sizes, int n_in,
                              void* d_out, int out_size, void* d_ws, size_t ws_size,
                              hipStream_t stream) {
    (void)in_sizes; (void)n_in; (void)out_size; (void)ws_size;
    const float* x    = (const float*)d_in[0];
    const float* inw  = (const float*)d_in[1];
    const float* inb  = (const float*)d_in[2];
    float* out = (float*)d_out;

    float* ws = (float*)d_ws;
    float* h_buf   = ws;                       // 8192*64
    float* xz_buf  = h_buf   + ROWS * DMODEL;  // 8192*256
    float* u_buf   = xz_buf  + ROWS * 256;     // 8192*128
    float* xdbl    = u_buf   + ROWS * DINNER;  // 8192*48
    float* dt_buf  = xdbl    + ROWS * XDLD;    // 8192*128
    float* y_buf   = dt_buf  + ROWS * DINNER;  // 8192*128
    float* acc_buf = y_buf   + ROWS * DINNER;  // 8192*64
    float* Hend    = acc_buf + ROWS * DMODEL;  // 2*128*64*16
    float* Hin     = Hend    + BSZ * DINNER * NCHUNK * DSTATE;
    float* dtsum   = Hin     + BSZ * DINNER * NCHUNK * DSTATE; // 2*128*64

    // ---- input embedding ----
    embed_kernel<<<(ROWS * DMODEL + 255) / 256, 256, 0, stream>>>(x, inw, inb, h_buf);

    auto gemm_blocks = [](int M, int N) {
        int tiles = (M >> 4) * ((N + 15) >> 4);
        return (tiles + 3) / 4;
    };

    for (int l = 0; l < NLAYERS; ++l) {
        int lbase = 3 + 20 * l;
        for (int dir = 0; dir < 2; ++dir) {
            int mb = lbase + dir * 9;
            const float* m_inw   = (const float*)d_in[mb + 0];
            const float* m_convw = (const float*)d_in[mb + 1];
            const float* m_convb = (const float*)d_in[mb + 2];
            const float* m_xprj  = (const float*)d_in[mb + 3];
            const float* m_dtw   = (const float*)d_in[mb + 4];
            const float* m_dtb   = (const float*)d_in[mb + 5];
            const float* m_alog  = (const float*)d_in[mb + 6];
            const float* m_D     = (const float*)d_in[mb + 7];
            const float* m_outw  = (const float*)d_in[mb + 8];
            int rev = dir;

            // xz = h @ in_w^T            (8192x64 @ 64x256), K=64
            gemm_wmma_f32<64, 0, 0><<<gemm_blocks(ROWS, 256), 128, 0, stream>>>(
                h_buf, DMODEL, m_inw, nullptr, xz_buf, 256, ROWS, 256);

            // u = silu(depthwise_conv(xz[:, :128]))
            conv_silu_kernel<<<(ROWS * DINNER + 255) / 256, 256, 0, stream>>>(
                xz_buf, m_convw, m_convb, u_buf, rev);

            // x_dbl = u @ xproj_w^T      (8192x128 @ 128x36, ldc padded 48), K=128
            gemm_wmma_f32<128, 0, 0><<<gemm_blocks(ROWS, 36), 128, 0, stream>>>(
                u_buf, DINNER, m_xprj, nullptr, xdbl, XDLD, ROWS, 36);

            // dt = softplus(x_dbl[:, :4] @ dt_w^T + dt_b)  (K=4 -> one WMMA)
            gemm_wmma_f32<4, 1, 0><<<gemm_blocks(ROWS, DINNER), 128, 0, stream>>>(
                xdbl, XDLD, m_dtw, m_dtb, dt_buf, DINNER, ROWS, DINNER);

            // chunked selective scan
            scan_phase1<<<(BSZ * DINNER * NCHUNK + 127) / 128, 128, 0, stream>>>(
                dt_buf, u_buf, xdbl, m_alog, Hend, dtsum, rev);
            scan_phase2<<<1, BSZ * DINNER, 0, stream>>>(
                Hend, dtsum, m_alog, Hin);
            scan_phase3<<<(BSZ * DINNER * NCHUNK + 127) / 128, 128, 0, stream>>>(
                dt_buf, u_buf, xdbl, xz_buf, m_alog, m_D, Hin, y_buf, rev);

            // acc (+)= y @ out_w^T       (8192x128 @ 128x64), K=128
            if (dir == 0) {
                gemm_wmma_f32<128, 0, 0><<<gemm_blocks(ROWS, DMODEL), 128, 0, stream>>>(
                    y_buf, DINNER, m_outw, nullptr, acc_buf, DMODEL, ROWS, DMODEL);
            } else {
                gemm_wmma_f32<128, 0, 1><<<gemm_blocks(ROWS, DMODEL), 128, 0, stream>>>(
                    y_buf, DINNER, m_outw, nullptr, acc_buf, DMODEL, ROWS, DMODEL);
            }
        }
        // h = LN(out_f + out_b + h)
        const float* ln_g = (const float*)d_in[lbase + 18];
        const float* ln_b = (const float*)d_in[lbase + 19];
        add_ln_kernel<<<(ROWS + 255) / 256, 256, 0, stream>>>(
            h_buf, acc_buf, ln_g, ln_b);
    }

    // final norm + head
    const float* norm_g = (const float*)d_in[83];
    const float* norm_b = (const float*)d_in[84];
    const float* head_w = (const float*)d_in[85];
    const float* head_b = (const float*)d_in[86];
    add_ln_kernel<<<(ROWS + 255) / 256, 256, 0, stream>>>(
        h_buf, nullptr, norm_g, norm_b);
    head_kernel<<<(ROWS * NCLS + 255) / 256, 256, 0, stream>>>(
        h_buf, head_w, head_b, out);
}